// GATLayer_58437325029857
// MI455X (gfx1250) — compile-verified
//
#include <hip/hip_runtime.h>
#include <hip/hip_bf16.h>

#define D_IN   128
#define D_OUT  64
#define ALPHA  0.2f
#define EPS    1e-10f

typedef __attribute__((ext_vector_type(2))) float v2f;
typedef __attribute__((ext_vector_type(8))) float v8f;

// ---- float <-> order-preserving unsigned (for atomic segment-max) ----
__device__ __forceinline__ unsigned f2ord(float f) {
    unsigned u = __float_as_uint(f);
    return (u & 0x80000000u) ? ~u : (u | 0x80000000u);
}
__device__ __forceinline__ float ord2f(unsigned u) {
    return __uint_as_float((u & 0x80000000u) ? (u ^ 0x80000000u) : ~u);
}
#define ORD_NEGINF 0x007FFFFFu   // f2ord(-inf)

__device__ __forceinline__ float leaky(float x) { return x > 0.f ? x : ALPHA * x; }

// ---------------------------------------------------------------------
// 1) init: zero out[N*64], denom[N]; m_flip[N] = flip(-inf)
// ---------------------------------------------------------------------
__global__ void gat_init(float* __restrict__ out, float* __restrict__ denom,
                         unsigned* __restrict__ mflip, int N) {
    int i = blockIdx.x * blockDim.x + threadIdx.x;
    int total = N * D_OUT;
    if (i < total) out[i] = 0.0f;
    if (i < N) { denom[i] = 0.0f; mflip[i] = ORD_NEGINF; }
}

// ---------------------------------------------------------------------
// 2) GEMM h = node @ W via V_WMMA_F32_16X16X4_F32.
//    One wave handles 16 rows x 64 cols (four 16x16 C tiles), K=128.
//    ISA f32 layouts (CDNA5 ISA §7.12.2):
//      A 16x4 : VGPR0/1, lanes 0-15 -> K=0/1, lanes 16-31 -> K=2/3, M=lane&15
//      B 4x16 : VGPR0/1, lanes 0-15 -> K=0/1, lanes 16-31 -> K=2/3, N=lane&15
//      C 16x16: VGPR r, lanes 0-15 -> M=r, lanes 16-31 -> M=r+8, N=lane&15
// ---------------------------------------------------------------------
__global__ void gat_gemm_wmma(const float* __restrict__ node,
                              const float* __restrict__ W,
                              float* __restrict__ h, int N) {
    int wave = (blockIdx.x * blockDim.x + threadIdx.x) >> 5;
    int lane = threadIdx.x & 31;
    int row0 = wave * 16;
    if (row0 >= N) return;

    int mrow = lane & 15;          // M (and N for B/C) within tile
    int koff = (lane >> 4) * 2;    // upper half-wave covers K+2,K+3

    v8f c0 = {}, c1 = {}, c2 = {}, c3 = {};

    const float* arow = node + (size_t)(row0 + mrow) * D_IN + koff;

    for (int k = 0; k < D_IN; k += 4) {
        v2f a;
        a.x = arow[k];
        a.y = arow[k + 1];

        const float* bp = W + (size_t)(k + koff) * D_OUT + mrow;
        v2f b0, b1, b2, b3;
        b0.x = bp[0];            b0.y = bp[D_OUT];
        b1.x = bp[16];           b1.y = bp[16 + D_OUT];
        b2.x = bp[32];           b2.y = bp[32 + D_OUT];
        b3.x = bp[48];           b3.y = bp[48 + D_OUT];

        c0 = __builtin_amdgcn_wmma_f32_16x16x4_f32(false, a, false, b0, (short)0, c0, false, false);
        c1 = __builtin_amdgcn_wmma_f32_16x16x4_f32(false, a, false, b1, (short)0, c1, false, false);
        c2 = __builtin_amdgcn_wmma_f32_16x16x4_f32(false, a, false, b2, (short)0, c2, false, false);
        c3 = __builtin_amdgcn_wmma_f32_16x16x4_f32(false, a, false, b3, (short)0, c3, false, false);
    }

    int nlo   = lane & 15;
    int rbase = row0 + ((lane >> 4) * 8);
#pragma unroll
    for (int r = 0; r < 8; ++r) {
        float* op = h + (size_t)(rbase + r) * D_OUT + nlo;
        op[0]  = c0[r];
        op[16] = c1[r];
        op[32] = c2[r];
        op[48] = c3[r];
    }
}

// ---------------------------------------------------------------------
// 3) per-node attention scores: ssrc = h . a[:64], sdst = h . a[64:]
// ---------------------------------------------------------------------
__global__ void gat_scores(const float* __restrict__ h, const float* __restrict__ a,
                           float* __restrict__ ssrc, float* __restrict__ sdst, int N) {
    int i = blockIdx.x * blockDim.x + threadIdx.x;
    if (i >= N) return;
    const float* hr = h + (size_t)i * D_OUT;
    float s0 = 0.f, s1 = 0.f;
#pragma unroll 8
    for (int j = 0; j < D_OUT; ++j) {
        float v = hr[j];
        s0 = fmaf(v, a[j], s0);
        s1 = fmaf(v, a[D_OUT + j], s1);
    }
    ssrc[i] = s0;
    sdst[i] = s1;
}

// ---------------------------------------------------------------------
// 4) per-edge logit + segment max via ordered-uint atomicMax
// ---------------------------------------------------------------------
__global__ void gat_edge_max(const int* __restrict__ src, const int* __restrict__ dst,
                             const float* __restrict__ ssrc, const float* __restrict__ sdst,
                             unsigned* __restrict__ mflip, float* __restrict__ logits, int E) {
    int e = blockIdx.x * blockDim.x + threadIdx.x;
    if (e >= E) return;
    int s = src[e], d = dst[e];
    float l = leaky(ssrc[s] + sdst[d]);
    logits[e] = l;
    atomicMax(&mflip[s], f2ord(l));
}

// ---------------------------------------------------------------------
// 5) finalize m: unflip; -inf -> 0 (matches reference's where())
// ---------------------------------------------------------------------
__global__ void gat_m_final(const unsigned* __restrict__ mflip, float* __restrict__ m, int N) {
    int i = blockIdx.x * blockDim.x + threadIdx.x;
    if (i >= N) return;
    float v = ord2f(mflip[i]);
    m[i] = __isinff(v) && v < 0.f ? 0.0f : v;
}

// ---------------------------------------------------------------------
// 6) per-edge exp + segment sum (native global_atomic_add_f32, L2 resident)
// ---------------------------------------------------------------------
__global__ void gat_edge_exp(const int* __restrict__ src, const float* __restrict__ logits,
                             const float* __restrict__ m, float* __restrict__ denom, int E) {
    int e = blockIdx.x * blockDim.x + threadIdx.x;
    if (e >= E) return;
    int s = src[e];
    atomicAdd(&denom[s], __expf(logits[e] - m[s]));
}

// ---------------------------------------------------------------------
// 7) weighted scatter: one wave32 per edge, lane covers 2 features.
//    Coalesced 256B gather of h[dst]; 2 f32 atomics/lane into out[src].
// ---------------------------------------------------------------------
__global__ void gat_edge_scatter(const int* __restrict__ src, const int* __restrict__ dst,
                                 const float* __restrict__ logits, const float* __restrict__ m,
                                 const float* __restrict__ denom, const float* __restrict__ h,
                                 float* __restrict__ out, int E) {
    long long gid = (long long)blockIdx.x * blockDim.x + threadIdx.x;
    int e    = (int)(gid >> 5);
    int lane = (int)(gid & 31);
    if (e >= E) return;
    int s = src[e], d = dst[e];
    float att = __expf(logits[e] - m[s]) / (denom[s] + EPS);
    const float2* hp = (const float2*)(h + (size_t)d * D_OUT);
    float2 v = hp[lane];
    float* op = out + (size_t)s * D_OUT + 2 * lane;
    atomicAdd(op + 0, att * v.x);
    atomicAdd(op + 1, att * v.y);
}

// ---------------------------------------------------------------------
extern "C" void kernel_launch(void* const* d_in, const int* in_sizes, int n_in,
                              void* d_out, int out_size, void* d_ws, size_t ws_size,
                              hipStream_t stream) {
    const float* node = (const float*)d_in[0];
    const int*   ei   = (const int*)d_in[1];
    const float* W    = (const float*)d_in[2];
    const float* a    = (const float*)d_in[3];
    float* out        = (float*)d_out;

    int N = in_sizes[0] / D_IN;
    int E = in_sizes[1] / 2;
    const int* src = ei;
    const int* dst = ei + E;

    // workspace layout
    float*    h      = (float*)d_ws;                  // N*64
    float*    ssrc   = h + (size_t)N * D_OUT;         // N
    float*    sdst   = ssrc + N;                      // N
    float*    m      = sdst + N;                      // N
    unsigned* mflip  = (unsigned*)(m + N);            // N
    float*    denom  = (float*)(mflip + N);           // N
    float*    logits = denom + N;                     // E

    const int B = 256;

    // 1) init
    {
        int total = N * D_OUT;
        gat_init<<<(total + B - 1) / B, B, 0, stream>>>(out, denom, mflip, N);
    }
    // 2) GEMM (WMMA f32): one wave per 16 rows
    {
        int waves   = (N + 15) / 16;
        long long threads = (long long)waves * 32;
        int blocks  = (int)((threads + B - 1) / B);
        gat_gemm_wmma<<<blocks, B, 0, stream>>>(node, W, h, N);
    }
    // 3) scores
    gat_scores<<<(N + B - 1) / B, B, 0, stream>>>(h, a, ssrc, sdst, N);
    // 4) per-edge max
    gat_edge_max<<<(E + B - 1) / B, B, 0, stream>>>(src, dst, ssrc, sdst, mflip, logits, E);
    // 5) finalize m
    gat_m_final<<<(N + B - 1) / B, B, 0, stream>>>(mflip, m, N);
    // 6) per-edge exp-sum
    gat_edge_exp<<<(E + B - 1) / B, B, 0, stream>>>(src, logits, m, denom, E);
    // 7) weighted scatter, wave per edge
    {
        long long threads = (long long)E * 32;
        int blocks = (int)((threads + B - 1) / B);
        gat_edge_scatter<<<blocks, B, 0, stream>>>(src, dst, logits, m, denom, h, out, E);
    }
}